// VQVAE_11038065951551
// MI455X (gfx1250) — compile-verified
//
#include <hip/hip_runtime.h>
#include <hip/hip_bf16.h>

typedef float v2f __attribute__((ext_vector_type(2)));
typedef float v8f __attribute__((ext_vector_type(8)));

#define N_TOT    131072
#define M_BLOCK  64
#define NBLOCKS  (N_TOT / M_BLOCK)   /* 2048 */
#define NTHREADS 256

#define VQ_OFF   ((size_t)N_TOT * 128)       /* 16777216 */
#define REC_OFF  (VQ_OFF + 1)
#define IDX_OFF  (VQ_OFF + 2)

__device__ __forceinline__ v8f wmma4(v2f a, v2f b, v8f c) {
  // D = A(16x4,f32) x B(4x16,f32) + C(16x16,f32)
  return __builtin_amdgcn_wmma_f32_16x16x4_f32(
      /*neg_a=*/false, a, /*neg_b=*/false, b,
      /*c_mod=*/(short)0, c, /*reuse_a=*/false, /*reuse_b=*/false);
}

// One MLP layer over the block's 64-row tile.
// W: [din x dout] row-major, B: [dout], hin: LDS [64 x din], hout: LDS [64 x dout].
__device__ __forceinline__ void mlp_layer(
    const float* __restrict__ W, const float* __restrict__ B,
    const float* __restrict__ hin, float* __restrict__ hout,
    const int din, const int dout, const bool relu)
{
  const int lane = threadIdx.x & 31;
  const int wave = threadIdx.x >> 5;
  const int hi   = lane >> 4;   // 0: K=4k..4k+1, 1: K=4k+2..4k+3
  const int lo   = lane & 15;
  const int ntiles = dout >> 4;
  const int ksteps = din >> 2;

  for (int nt = wave; nt < ntiles; nt += 8) {
    const int ncol = (nt << 4) + lo;
    v8f acc[4];
#pragma unroll
    for (int m = 0; m < 4; ++m)
      acc[m] = v8f{0.f, 0.f, 0.f, 0.f, 0.f, 0.f, 0.f, 0.f};

    const float* __restrict__ wc = W + ncol;
#pragma unroll 2
    for (int k = 0; k < ksteps; ++k) {
      const int kk = (k << 2) + (hi << 1);
      v2f bf;
      bf.x = wc[(size_t)kk * dout];
      bf.y = wc[(size_t)(kk + 1) * dout];
#pragma unroll
      for (int m = 0; m < 4; ++m) {
        v2f af = *(const v2f*)(hin + (m * 16 + lo) * din + kk);
        acc[m] = wmma4(af, bf, acc[m]);
      }
    }
    const float bv = B[ncol];
#pragma unroll
    for (int m = 0; m < 4; ++m) {
#pragma unroll
      for (int r = 0; r < 8; ++r) {
        float v = acc[m][r] + bv;
        if (relu) v = fmaxf(v, 0.f);
        hout[(m * 16 + (hi << 3) + r) * dout + ncol] = v;
      }
    }
  }
}

// Final decoder layer 512->128: write x_hat to global and accumulate rec partial.
__device__ __forceinline__ void mlp_layer_out(
    const float* __restrict__ W, const float* __restrict__ B,
    const float* __restrict__ hin, const float* __restrict__ x,
    float* __restrict__ xhat, const int row0, float& recAcc)
{
  const int din = 512, dout = 128;
  const int lane = threadIdx.x & 31;
  const int wave = threadIdx.x >> 5;
  const int hi   = lane >> 4;
  const int lo   = lane & 15;
  const int nt   = wave;          // 8 tiles, 8 waves
  const int ncol = (nt << 4) + lo;

  v8f acc[4];
#pragma unroll
  for (int m = 0; m < 4; ++m)
    acc[m] = v8f{0.f, 0.f, 0.f, 0.f, 0.f, 0.f, 0.f, 0.f};

  const float* __restrict__ wc = W + ncol;
#pragma unroll 2
  for (int k = 0; k < din / 4; ++k) {
    const int kk = (k << 2) + (hi << 1);
    v2f bf;
    bf.x = wc[(size_t)kk * dout];
    bf.y = wc[(size_t)(kk + 1) * dout];
#pragma unroll
    for (int m = 0; m < 4; ++m) {
      v2f af = *(const v2f*)(hin + (m * 16 + lo) * din + kk);
      acc[m] = wmma4(af, bf, acc[m]);
    }
  }
  const float bv = B[ncol];
#pragma unroll
  for (int m = 0; m < 4; ++m) {
#pragma unroll
    for (int r = 0; r < 8; ++r) {
      const int grow = row0 + m * 16 + (hi << 3) + r;
      const size_t off = (size_t)grow * 128 + ncol;
      float v = acc[m][r] + bv;
      xhat[off] = v;
      float d = v - x[off];
      recAcc = fmaf(d, d, recAcc);
    }
  }
}

__global__ __launch_bounds__(NTHREADS, 1)
void vqvae_fused_kernel(
    const float* __restrict__ x,
    const float* __restrict__ We1, const float* __restrict__ be1,
    const float* __restrict__ We2, const float* __restrict__ be2,
    const float* __restrict__ We3, const float* __restrict__ be3,
    const float* __restrict__ We4, const float* __restrict__ be4,
    const float* __restrict__ Wd1, const float* __restrict__ bd1,
    const float* __restrict__ Wd2, const float* __restrict__ bd2,
    const float* __restrict__ Wd3, const float* __restrict__ bd3,
    const float* __restrict__ Wd4, const float* __restrict__ bd4,
    const float* __restrict__ cbk,
    float* __restrict__ out, float* __restrict__ ws)
{
  __shared__ float bufA[M_BLOCK * 512];   // 128 KB
  __shared__ float bufB[M_BLOCK * 512];   // 128 KB
  __shared__ float cb[128 * 32];          // 16 KB codebook
  __shared__ float red[NTHREADS];

  const int tid  = threadIdx.x;
  const int blk  = blockIdx.x;
  const int row0 = blk * M_BLOCK;

  // Stage x tile (64x128) and codebook into LDS (float4 vectorized).
  {
    const float4* xs = (const float4*)(x + (size_t)row0 * 128);
    float4* bs = (float4*)bufA;
    for (int i = tid; i < M_BLOCK * 128 / 4; i += NTHREADS) bs[i] = xs[i];
    const float4* cs = (const float4*)cbk;
    float4* cd = (float4*)cb;
    for (int i = tid; i < 128 * 32 / 4; i += NTHREADS) cd[i] = cs[i];
  }
  __syncthreads();

  float vqAcc = 0.f, recAcc = 0.f;

  // Encoder
  mlp_layer(We1, be1, bufA, bufB, 128, 512, true);  __syncthreads();
  mlp_layer(We2, be2, bufB, bufA, 512, 256, true);  __syncthreads();
  mlp_layer(We3, be3, bufA, bufB, 256, 64,  true);  __syncthreads();
  mlp_layer(We4, be4, bufB, bufA, 64,  32,  false); __syncthreads();

  // VQ: z in bufA [64x32] -> z_q in bufB, indices + vq partial.
  {
    const int row = tid >> 2;     // 0..63
    const int q   = tid & 3;      // 4 threads per row, contiguous code chunks
    float zr[32];
#pragma unroll
    for (int d = 0; d < 32; ++d) zr[d] = bufA[row * 32 + d];
    float bestd = 3.402823e38f;
    int   besti = 1 << 30;
    for (int c = q * 32; c < q * 32 + 32; ++c) {
      float s = 0.f;
#pragma unroll
      for (int d = 0; d < 32; ++d) {
        float t = zr[d] - cb[c * 32 + d];
        s = fmaf(t, t, s);
      }
      if (s < bestd) { bestd = s; besti = c; }
    }
#pragma unroll
    for (int off = 1; off < 4; off <<= 1) {
      float od = __shfl_xor(bestd, off, 32);
      int   oi = __shfl_xor(besti, off, 32);
      if (od < bestd || (od == bestd && oi < besti)) { bestd = od; besti = oi; }
    }
    if (q == 0) {
      out[IDX_OFF + (size_t)(row0 + row)] = (float)besti;
      float s = 0.f;
#pragma unroll
      for (int d = 0; d < 32; ++d) {
        float qv = cb[besti * 32 + d];
        bufB[row * 32 + d] = qv;            // straight-through: decoder sees z_q
        float t = qv - zr[d];
        s = fmaf(t, t, s);
      }
      vqAcc = s;
    }
  }
  __syncthreads();

  // Decoder
  mlp_layer(Wd1, bd1, bufB, bufA, 32,  64,  true);  __syncthreads();
  mlp_layer(Wd2, bd2, bufA, bufB, 64,  256, true);  __syncthreads();
  mlp_layer(Wd3, bd3, bufB, bufA, 256, 512, true);  __syncthreads();
  mlp_layer_out(Wd4, bd4, bufA, x, out, row0, recAcc);

  // Block-level reduction of loss partials -> d_ws (deterministic).
  red[tid] = vqAcc;  __syncthreads();
  for (int s = NTHREADS / 2; s > 0; s >>= 1) {
    if (tid < s) red[tid] += red[tid + s];
    __syncthreads();
  }
  float vqSum = red[0];
  __syncthreads();
  red[tid] = recAcc; __syncthreads();
  for (int s = NTHREADS / 2; s > 0; s >>= 1) {
    if (tid < s) red[tid] += red[tid + s];
    __syncthreads();
  }
  if (tid == 0) { ws[2 * blk] = vqSum; ws[2 * blk + 1] = red[0]; }
}

__global__ __launch_bounds__(256, 1)
void vqvae_reduce_kernel(const float* __restrict__ ws, float* __restrict__ out)
{
  __shared__ float sv[256], sr[256];
  float v = 0.f, r = 0.f;
  for (int i = threadIdx.x; i < NBLOCKS; i += 256) {
    v += ws[2 * i];
    r += ws[2 * i + 1];
  }
  sv[threadIdx.x] = v; sr[threadIdx.x] = r;
  __syncthreads();
  for (int s = 128; s > 0; s >>= 1) {
    if (threadIdx.x < s) {
      sv[threadIdx.x] += sv[threadIdx.x + s];
      sr[threadIdx.x] += sr[threadIdx.x + s];
    }
    __syncthreads();
  }
  if (threadIdx.x == 0) {
    // vq_loss = codebook + beta*commitment = 2 * mean((z_q - z)^2)
    out[VQ_OFF]  = 2.0f * sv[0] / (float)((size_t)N_TOT * 32);
    out[REC_OFF] = sr[0] / (float)((size_t)N_TOT * 128);
  }
}

extern "C" void kernel_launch(void* const* d_in, const int* in_sizes, int n_in,
                              void* d_out, int out_size, void* d_ws, size_t ws_size,
                              hipStream_t stream) {
  (void)in_sizes; (void)n_in; (void)out_size; (void)ws_size;
  // setup_inputs() dict order: x, WeW1,Web1..WeW4,Web4, WdW1,Wdb1..WdW4,Wdb4, codebook
  const float* x   = (const float*)d_in[0];
  const float* We1 = (const float*)d_in[1];  const float* be1 = (const float*)d_in[2];
  const float* We2 = (const float*)d_in[3];  const float* be2 = (const float*)d_in[4];
  const float* We3 = (const float*)d_in[5];  const float* be3 = (const float*)d_in[6];
  const float* We4 = (const float*)d_in[7];  const float* be4 = (const float*)d_in[8];
  const float* Wd1 = (const float*)d_in[9];  const float* bd1 = (const float*)d_in[10];
  const float* Wd2 = (const float*)d_in[11]; const float* bd2 = (const float*)d_in[12];
  const float* Wd3 = (const float*)d_in[13]; const float* bd3 = (const float*)d_in[14];
  const float* Wd4 = (const float*)d_in[15]; const float* bd4 = (const float*)d_in[16];
  const float* cbk = (const float*)d_in[17];

  float* out = (float*)d_out;
  float* ws  = (float*)d_ws;

  vqvae_fused_kernel<<<NBLOCKS, NTHREADS, 0, stream>>>(
      x, We1, be1, We2, be2, We3, be3, We4, be4,
      Wd1, bd1, Wd2, bd2, Wd3, bd3, Wd4, bd4, cbk, out, ws);
  vqvae_reduce_kernel<<<1, 256, 0, stream>>>(ws, out);
}